// FlowLoss_28836410425931
// MI455X (gfx1250) — compile-verified
//
#include <hip/hip_runtime.h>
#include <hip/hip_bf16.h>
#include <math.h>

#define N_LVL 5
#define B_    16
#define H_    384
#define W_    384
#define J_    17
#define NKP   (B_ * J_)          // 272
#define TH    8                  // rows per tile
#define NTILE (H_ / TH)          // 48
#define NPART (N_LVL * B_ * NTILE) // 3840
#define HW    (H_ * W_)
#define GAMMA 0.8f

typedef __attribute__((ext_vector_type(2))) float v2f;
typedef __attribute__((ext_vector_type(8))) float v8f;

__device__ __forceinline__ float charb(float d) {
    // (d^2 + eps^2)^0.25, eps = 1e-9  ->  sqrt(sqrt(.))
    return sqrtf(sqrtf(fmaf(d, d, 1e-18f)));
}

// ---------------------------------------------------------------------------
// Supervised EPE term: gt flow is a sparse scatter of <=272 keypoints.
// Single block, deterministic tree reductions. Writes sup loss to supOut[0].
// ---------------------------------------------------------------------------
__global__ __launch_bounds__(256)
void flowloss_sup_kernel(const float* __restrict__ pf,
                         const float* __restrict__ kps,
                         float* __restrict__ supOut) {
    __shared__ int   key[NKP];
    __shared__ float kdx[NKP], kdy[NKP];
    __shared__ int   mflag[NKP];
    __shared__ int   cnt;
    __shared__ float red[256];
    const int tid = threadIdx.x;
    if (tid == 0) cnt = 0;
    __syncthreads();

    // build scatter entries
    for (int e = tid; e < NKP; e += 256) {
        int b = e / J_, j = e % J_;
        const float* kp = kps + (size_t)b * (2 * J_ * 2);
        float x0 = kp[j * 2 + 0],          y0 = kp[j * 2 + 1];
        float x1 = kp[J_ * 2 + j * 2 + 0], y1 = kp[J_ * 2 + j * 2 + 1];
        bool valid = (x0 >= 0.f) && (y0 >= 0.f) && (x1 >= 0.f) && (y1 >= 0.f) &&
                     (x0 < (float)W_) && (y0 < (float)H_) &&
                     (x1 < (float)W_) && (y1 < (float)H_);
        if (valid) {
            int xi = (int)floorf(x0), yi = (int)floorf(y0);
            key[e] = b * HW + yi * W_ + xi;      // globally unique pixel id
            kdx[e] = x1 - x0;
            kdy[e] = y1 - y0;
        } else {
            key[e] = -1;
            kdx[e] = 0.f; kdy[e] = 0.f;
        }
    }
    __syncthreads();

    // dedupe (last scatter wins) + mask (|disp| > 0) + count
    for (int e = tid; e < NKP; e += 256) {
        bool surv = key[e] >= 0;
        if (surv) {
            for (int e2 = e + 1; e2 < NKP; ++e2)
                if (key[e2] == key[e]) { surv = false; break; }
        }
        bool m = surv && (kdx[e] * kdx[e] + kdy[e] * kdy[e] > 0.f);
        mflag[e] = m ? 1 : 0;
        if (m) atomicAdd(&cnt, 1);
    }
    __syncthreads();

    // per-level masked EPE sums (gathers: 5 * 272 * 2 loads total)
    float lsum[N_LVL];
    #pragma unroll
    for (int i = 0; i < N_LVL; ++i) lsum[i] = 0.f;
    for (int e = tid; e < NKP; e += 256) {
        if (!mflag[e]) continue;
        int b   = e / J_;
        int pix = key[e] - b * HW;
        #pragma unroll
        for (int i = 0; i < N_LVL; ++i) {
            const float* base = pf + ((size_t)(i * B_ + b) * 2) * HW;
            float du = base[pix]      - kdx[e];
            float dv = base[HW + pix] - kdy[e];
            lsum[i] += sqrtf(du * du + dv * dv);
        }
    }

    float sup = 0.f;
    #pragma unroll
    for (int i = 0; i < N_LVL; ++i) {
        red[tid] = lsum[i];
        __syncthreads();
        for (int s = 128; s > 0; s >>= 1) {
            if (tid < s) red[tid] += red[tid + s];
            __syncthreads();
        }
        if (tid == 0) sup += powf(GAMMA, (float)(N_LVL - 1 - i)) * red[0];
        __syncthreads();
    }
    if (tid == 0) {
        float denom = fmaxf((float)cnt, 1.f);
        supOut[0] = sup / denom;   // LOSS_WEIGHT == 1
    }
}

// ---------------------------------------------------------------------------
// Unsupervised term (photometric warp + smoothness), fused per (level,b,tile).
// Flow tile (TH+1 rows x 384 x 2ch) staged to LDS via async global->LDS
// copies (ASYNCcnt path) so the 3x stencil/warp reuse never re-touches cache.
// One deterministic partial per block -> ws[partial_id].
// ---------------------------------------------------------------------------
__global__ __launch_bounds__(256)
void flowloss_unsup_kernel(const float* __restrict__ pf,
                           const float* __restrict__ frm0,
                           const float* __restrict__ frm1,
                           float* __restrict__ partials) {
    __shared__ float smem[2 * (TH + 1) * W_];   // 27648 B
    __shared__ float wsum[8];

    const int tid = threadIdx.x;
    const int bx  = blockIdx.x;       // tile
    const int b   = blockIdx.y;       // batch
    const int lvl = blockIdx.z;       // pyramid level
    const int y0  = bx * TH;

    // ---- stage flow tile: rows y0..y0+TH, 2 channels, 96 x b128 per row ----
    const int CHUNKS = (TH + 1) * 2 * (W_ / 4);   // 1728
    for (int t = tid; t < CHUNKS; t += 256) {
        int r   = t / (2 * (W_ / 4));
        int rem = t % (2 * (W_ / 4));
        int c   = rem / (W_ / 4);
        int q   = rem % (W_ / 4);
        int yy  = y0 + r;
        int lidx = (c * (TH + 1) + r) * W_ + q * 4;
        if (yy < H_) {
            const float* g = pf + (((size_t)(lvl * B_ + b) * 2 + c) * H_ + yy) * W_ + q * 4;
            unsigned ldsOff = (unsigned)(lidx * 4);   // byte offset from LDS base
            asm volatile("global_load_async_to_lds_b128 %0, %1, off"
                         :: "v"(ldsOff), "v"(g) : "memory");
        } else {
            float4 z = make_float4(0.f, 0.f, 0.f, 0.f);
            *(float4*)&smem[lidx] = z;
        }
    }
    asm volatile("s_wait_asynccnt 0" ::: "memory");
    __syncthreads();

    const float* fx_p = smem;                    // channel 0 plane
    const float* fy_p = smem + (TH + 1) * W_;    // channel 1 plane
    const float sx = (float)(W_ - 1) / (float)W_;
    const float sy = (float)(H_ - 1) / (float)H_;

    float acc = 0.f;
    for (int p = tid; p < TH * W_; p += 256) {
        int ly = p / W_, x = p % W_;
        int y  = y0 + ly;

        float fx = fx_p[ly * W_ + x];
        float fy = fy_p[ly * W_ + x];

        // smoothness: forward diffs with zero pad at bottom/right
        float fxv = fx_p[(ly + 1) * W_ + x];
        float fyv = fy_p[(ly + 1) * W_ + x];
        float fxh = (x + 1 < W_) ? fx_p[ly * W_ + x + 1] : 0.f;
        float fyh = (x + 1 < W_) ? fy_p[ly * W_ + x + 1] : 0.f;
        float s = 0.5f * (charb(fx - fxv) + charb(fx - fxh) +
                          charb(fy - fyv) + charb(fy - fyh));

        // warp coords: xp = (fx+x)*(W-1)/W ; yp = (fy+y)*(H-1)/H
        float xp = (fx + (float)x) * sx;
        float yp = (fy + (float)y) * sy;
        float x0f = floorf(xp), y0f = floorf(yp);
        float wx1 = xp - x0f, wx0 = 1.f - wx1;
        float wy1 = yp - y0f, wy0 = 1.f - wy1;
        bool vx0 = (x0f >= 0.f) && (x0f <= (float)(W_ - 1));
        bool vx1 = (x0f + 1.f >= 0.f) && (x0f + 1.f <= (float)(W_ - 1));
        bool vy0 = (y0f >= 0.f) && (y0f <= (float)(H_ - 1));
        bool vy1 = (y0f + 1.f >= 0.f) && (y0f + 1.f <= (float)(H_ - 1));
        float w00 = wx0 * wy0 * (float)(vx0 && vy0);
        float w10 = wx1 * wy0 * (float)(vx1 && vy0);
        float w01 = wx0 * wy1 * (float)(vx0 && vy1);
        float w11 = wx1 * wy1 * (float)(vx1 && vy1);
        int xc0 = (int)fminf(fmaxf(x0f,       0.f), (float)(W_ - 1));
        int xc1 = (int)fminf(fmaxf(x0f + 1.f, 0.f), (float)(W_ - 1));
        int yc0 = (int)fminf(fmaxf(y0f,       0.f), (float)(H_ - 1));
        int yc1 = (int)fminf(fmaxf(y0f + 1.f, 0.f), (float)(H_ - 1));

        float pc = 0.f;
        #pragma unroll
        for (int c = 0; c < 3; ++c) {
            const float* im = frm1 + ((size_t)b * 3 + c) * HW;
            float samp = w00 * im[yc0 * W_ + xc0] + w10 * im[yc0 * W_ + xc1] +
                         w01 * im[yc1 * W_ + xc0] + w11 * im[yc1 * W_ + xc1];
            float d = samp - frm0[((size_t)b * 3 + c) * HW + y * W_ + x];
            pc += charb(d);
        }
        acc += s + pc * (1.f / 3.f);
    }

    // deterministic block reduce: wave32 shuffle tree, then 8 fixed-order adds
    #pragma unroll
    for (int o = 16; o > 0; o >>= 1) acc += __shfl_down(acc, o, 32);
    if ((tid & 31) == 0) wsum[tid >> 5] = acc;
    __syncthreads();
    if (tid == 0) {
        float tot = 0.f;
        #pragma unroll
        for (int w = 0; w < 8; ++w) tot += wsum[w];
        float wgt = powf(GAMMA, (float)(N_LVL - 1 - lvl));
        partials[(size_t)((lvl * B_ + b) * NTILE + bx)] = tot * (wgt / (float)B_);
    }
}

// ---------------------------------------------------------------------------
// Final reduction: one wave32. 64 per-lane accumulator slots form the 16x4
// f32 A-matrix; v_wmma_f32_16x16x4_f32 against all-ones B does the row sums.
// ---------------------------------------------------------------------------
__global__ __launch_bounds__(32)
void flowloss_final_kernel(const float* __restrict__ partials,
                           const float* __restrict__ supSlot,
                           float* __restrict__ out) {
    const int lane = threadIdx.x;
    v2f a; a.x = 0.f; a.y = 0.f;
    for (int j = lane; j < NPART; j += 64)      a.x += partials[j];
    for (int j = lane + 32; j < NPART; j += 64) a.y += partials[j];

#if __has_builtin(__builtin_amdgcn_wmma_f32_16x16x4_f32)
    v2f bones; bones.x = 1.f; bones.y = 1.f;
    v8f c = {};
    v8f d = __builtin_amdgcn_wmma_f32_16x16x4_f32(
        /*neg_a=*/false, a, /*neg_b=*/false, bones,
        /*c_mod=*/(short)0, c, /*reuse_a=*/false, /*reuse_b=*/false);
    // D[:,0] (row sums) live in lane 0 (M=0..7) and lane 16 (M=8..15)
    float rs = d[0] + d[1] + d[2] + d[3] + d[4] + d[5] + d[6] + d[7];
    float total = __shfl(rs, 0, 32) + __shfl(rs, 16, 32);
#else
    float rs = a.x + a.y;
    #pragma unroll
    for (int o = 16; o > 0; o >>= 1) rs += __shfl_down(rs, o, 32);
    float total = __shfl(rs, 0, 32);
#endif
    if (lane == 0) out[0] = total + supSlot[0];
}

// ---------------------------------------------------------------------------
extern "C" void kernel_launch(void* const* d_in, const int* in_sizes, int n_in,
                              void* d_out, int out_size, void* d_ws, size_t ws_size,
                              hipStream_t stream) {
    const float* pf   = (const float*)d_in[0];   // (5,16,2,384,384)
    const float* kps  = (const float*)d_in[1];   // (16,2,17,2)
    const float* frm0 = (const float*)d_in[2];   // (16,3,384,384)
    const float* frm1 = (const float*)d_in[3];   // (16,3,384,384)
    float* ws  = (float*)d_ws;                   // [0..NPART) partials, [NPART] sup
    float* out = (float*)d_out;

    flowloss_sup_kernel<<<1, 256, 0, stream>>>(pf, kps, ws + NPART);

    dim3 grid(NTILE, B_, N_LVL);
    flowloss_unsup_kernel<<<grid, 256, 0, stream>>>(pf, frm0, frm1, ws);

    flowloss_final_kernel<<<1, 32, 0, stream>>>(ws, ws + NPART, out);
}